// ChebNet_64991445123461
// MI455X (gfx1250) — compile-verified
//
#include <hip/hip_runtime.h>
#include <hip/hip_bf16.h>

#define NN 50000
#define NE 1600000
#define NG 128

typedef __attribute__((ext_vector_type(2))) float v2f;
typedef __attribute__((ext_vector_type(8))) float v8f;

static __device__ __forceinline__ int gtid() {
    return blockIdx.x * blockDim.x + threadIdx.x;
}

// ---------------- utility kernels ----------------
__global__ void k_zero(float* p, int n) {
    int t = gtid();
    if (t < n) p[t] = 0.0f;
}

__global__ void k_degree(const int* __restrict__ src, const int* __restrict__ dst,
                         float* __restrict__ deg) {
    int e = gtid();
    if (e >= NE) return;
    int s = src[e], d = dst[e];
    if (s != d) atomicAdd(&deg[s], 1.0f);
}

__global__ void k_dinv(float* deg) {
    int n = gtid();
    if (n >= NN) return;
    float d = deg[n];
    deg[n] = (d > 0.0f) ? rsqrtf(d) : 0.0f;
}

__global__ void k_edgew(const int* __restrict__ src, const int* __restrict__ dst,
                        const float* __restrict__ dinv, float* __restrict__ w) {
    int e = gtid();
    if (e >= NE) return;
    int s = src[e], d = dst[e];
    w[e] = (s != d) ? (-dinv[s] * dinv[d]) : 0.0f;
}

// Tx[n][0..F) = h[n][0..F)   (Tx row stride = Kd = 5F)
__global__ void k_copy_tx0(const float* __restrict__ h, float* __restrict__ Tx,
                           int F, int Kd) {
    int t = gtid();
    if (t >= NN * F) return;
    int n = t / F, f = t - n * F;
    Tx[(size_t)n * Kd + f] = h[(size_t)n * F + f];
}

__global__ void k_zero_slice(float* __restrict__ Tx, int F, int Kd, int koff) {
    int t = gtid();
    if (t >= NN * F) return;
    int n = t / F, f = t - n * F;
    Tx[(size_t)n * Kd + koff + f] = 0.0f;
}

// Tx[:,k] = -Tx[:,k-2]
__global__ void k_init_neg(float* __restrict__ Tx, int F, int Kd, int kout, int kin) {
    int t = gtid();
    if (t >= NN * F) return;
    int n = t / F, f = t - n * F;
    Tx[(size_t)n * Kd + kout + f] = -Tx[(size_t)n * Kd + kin + f];
}

// Tx[:,kout][dst] += scale * w_e * Tx[:,kin][src]   (thread per edge x float4 group)
__global__ void k_scatter(const int* __restrict__ src, const int* __restrict__ dst,
                          const float* __restrict__ w, float* __restrict__ Tx,
                          int Kd, int kin, int kout, int F4, float scale) {
    int t = gtid();
    if (t >= NE * F4) return;
    int e = t / F4;
    int g = (t - e * F4) << 2;
    float we = w[e] * scale;
    if (we == 0.0f) return;
    int s = src[e], d = dst[e];
    const float4 z = *(const float4*)(Tx + (size_t)s * Kd + kin + g);
    float* o = Tx + (size_t)d * Kd + kout + g;
    atomicAdd(o + 0, we * z.x);
    atomicAdd(o + 1, we * z.y);
    atomicAdd(o + 2, we * z.z);
    atomicAdd(o + 3, we * z.w);
}

// C(N x O) = relu(A(N x Kd) @ B(Kd x O) + bias), one wave per 16x16 tile.
// Uses V_WMMA_F32_16X16X4_F32 (full f32 path); Kd, O compile-time so the
// K-loop has an exact trip count (no remainder path).
// A 16x4 layout: lane m=lane&15, VGPRs hold K = 2*(lane>>4) + {0,1}
// B 4x16 layout: lane n=lane&15, VGPRs hold K = 2*(lane>>4) + {0,1}
// C/D: VGPR r, lane L -> row = r + 8*(L>>4), col = L&15
template <int Kd, int O>
__global__ void k_gemm_bias_relu(const float* __restrict__ A, const float* __restrict__ B,
                                 const float* __restrict__ bias, float* __restrict__ C) {
    constexpr int Ot = O / 16;
    constexpr int nTiles = (NN / 16) * Ot;
    int wave = (blockIdx.x * blockDim.x + threadIdx.x) >> 5;
    if (wave >= nTiles) return;                 // wave-uniform: EXEC stays all-ones
    int mt = wave / Ot;
    int ot = wave - mt * Ot;
    int lane = threadIdx.x & 31;
    int m = mt * 16 + (lane & 15);
    int o = ot * 16 + (lane & 15);
    int kh = (lane >> 4) << 1;

    v8f c = {0.f, 0.f, 0.f, 0.f, 0.f, 0.f, 0.f, 0.f};
    const float* Arow = A + (size_t)m * Kd;
#pragma unroll 4
    for (int kk = 0; kk < Kd; kk += 4) {
        v2f a = *(const v2f*)(Arow + kk + kh);  // contiguous pair -> global_load_b64
        v2f b;
        b.x = B[(size_t)(kk + kh) * O + o];
        b.y = B[(size_t)(kk + kh + 1) * O + o];
        c = __builtin_amdgcn_wmma_f32_16x16x4_f32(false, a, false, b,
                                                  (short)0, c, false, false);
    }
    int r0 = mt * 16 + ((lane >> 4) << 3);
    float bb = bias[o];
#pragma unroll
    for (int r = 0; r < 8; ++r) {
        float v = c[r] + bb;
        C[(size_t)(r0 + r) * O + o] = v > 0.0f ? v : 0.0f;
    }
}

// ---------------- pooling + MLP ----------------
__global__ void k_pool_sum(const float* __restrict__ h, const int* __restrict__ batch,
                           float* __restrict__ pool) {
    int t = gtid();
    if (t >= NN * 64) return;
    int n = t >> 6, o = t & 63;
    atomicAdd(&pool[batch[n] * 64 + o], h[(size_t)n * 64 + o]);
}

__global__ void k_cnt(const int* __restrict__ batch, float* __restrict__ cnt) {
    int n = gtid();
    if (n >= NN) return;
    atomicAdd(&cnt[batch[n]], 1.0f);
}

// t1(128x32) = relu((pool/cnt) @ fcw1 + fcb1)
__global__ void k_fc1(const float* __restrict__ pool, const float* __restrict__ cnt,
                      const float* __restrict__ w, const float* __restrict__ b,
                      float* __restrict__ t1) {
    int t = gtid();
    if (t >= NG * 32) return;
    int g = t >> 5, o = t & 31;
    float s = 0.0f;
    for (int f = 0; f < 64; ++f) s += pool[g * 64 + f] * w[f * 32 + o];
    float c = cnt[g];
    c = c > 1.0f ? c : 1.0f;
    float v = s / c + b[o];
    t1[t] = v > 0.0f ? v : 0.0f;
}

__global__ void k_fc2(const float* __restrict__ t1, const float* __restrict__ w,
                      const float* __restrict__ b, float* __restrict__ out) {
    int g = gtid();
    if (g >= NG) return;
    float s = 0.0f;
    for (int o = 0; o < 32; ++o) s += t1[g * 32 + o] * w[o];
    out[g] = s + b[0];
}

// ---------------- host orchestration ----------------
static inline int G(long long n) { return (int)((n + 255) / 256); }

extern "C" void kernel_launch(void* const* d_in, const int* in_sizes, int n_in,
                              void* d_out, int out_size, void* d_ws, size_t ws_size,
                              hipStream_t stream) {
    const float* x      = (const float*)d_in[0];
    const int*   src    = (const int*)d_in[1];           // edge_index[0]
    const int*   dst    = ((const int*)d_in[1]) + NE;    // edge_index[1]
    const int*   batch  = (const int*)d_in[2];
    const float* W1 = (const float*)d_in[4];  const float* b1 = (const float*)d_in[5];
    const float* W2 = (const float*)d_in[6];  const float* b2 = (const float*)d_in[7];
    const float* W3 = (const float*)d_in[8];  const float* b3 = (const float*)d_in[9];
    const float* fcw1 = (const float*)d_in[10]; const float* fcb1 = (const float*)d_in[11];
    const float* fcw2 = (const float*)d_in[12]; const float* fcb2 = (const float*)d_in[13];
    float* out = (float*)d_out;

    // workspace layout (floats)
    float* ws   = (float*)d_ws;
    float* deg  = ws;                         // NN (becomes dinv in-place)
    float* ew   = deg + NN;                   // NE
    float* Tx   = ew + NE;                    // NN * 320 (max Kd)
    float* hA   = Tx + (size_t)NN * 320;      // NN * 64
    float* hB   = hA + (size_t)NN * 64;       // NN * 64
    float* pool = hB + (size_t)NN * 64;       // NG * 64
    float* cnt  = pool + NG * 64;             // NG
    float* t1   = cnt + NG;                   // NG * 32

    // graph normalization: deg -> dinv -> edge weights
    k_zero<<<G(NN), 256, 0, stream>>>(deg, NN);
    k_degree<<<G(NE), 256, 0, stream>>>(src, dst, deg);
    k_dinv<<<G(NN), 256, 0, stream>>>(deg);
    k_edgew<<<G(NE), 256, 0, stream>>>(src, dst, deg, ew);

    // Build the 5 Chebyshev bases for a layer into Tx (row layout [n][k][f], stride Kd=5F)
    auto build_tx = [&](const float* hin, int F) {
        int Kd = 5 * F;
        int F4 = F >> 2;
        k_copy_tx0<<<G((long long)NN * F), 256, 0, stream>>>(hin, Tx, F, Kd);
        k_zero_slice<<<G((long long)NN * F), 256, 0, stream>>>(Tx, F, Kd, F);
        k_scatter<<<G((long long)NE * F4), 256, 0, stream>>>(src, dst, ew, Tx, Kd, 0, F, F4, 1.0f);
        for (int k = 2; k < 5; ++k) {
            k_init_neg<<<G((long long)NN * F), 256, 0, stream>>>(Tx, F, Kd, k * F, (k - 2) * F);
            k_scatter<<<G((long long)NE * F4), 256, 0, stream>>>(src, dst, ew, Tx, Kd,
                                                                 (k - 1) * F, k * F, F4, 2.0f);
        }
    };

    // layer 1: F=4 -> O=32
    build_tx(x, 4);
    {
        constexpr int nT = (NN / 16) * (32 / 16);
        k_gemm_bias_relu<20, 32><<<G((long long)nT * 32), 256, 0, stream>>>(Tx, W1, b1, hA);
    }
    // layer 2: F=32 -> O=64
    build_tx(hA, 32);
    {
        constexpr int nT = (NN / 16) * (64 / 16);
        k_gemm_bias_relu<160, 64><<<G((long long)nT * 32), 256, 0, stream>>>(Tx, W2, b2, hB);
    }
    // layer 3: F=64 -> O=64
    build_tx(hB, 64);
    {
        constexpr int nT = (NN / 16) * (64 / 16);
        k_gemm_bias_relu<320, 64><<<G((long long)nT * 32), 256, 0, stream>>>(Tx, W3, b3, hA);
    }

    // global mean pool + MLP head
    k_zero<<<G(NG * 64), 256, 0, stream>>>(pool, NG * 64);
    k_zero<<<G(NG), 256, 0, stream>>>(cnt, NG);
    k_pool_sum<<<G((long long)NN * 64), 256, 0, stream>>>(hA, batch, pool);
    k_cnt<<<G(NN), 256, 0, stream>>>(batch, cnt);
    k_fc1<<<G(NG * 32), 256, 0, stream>>>(pool, cnt, fcw1, fcb1, t1);
    k_fc2<<<G(NG), 256, 0, stream>>>(t1, fcw2, fcb2, out);
}